// SAGE_73993696576014
// MI455X (gfx1250) — compile-verified
//
#include <hip/hip_runtime.h>

typedef float v2f __attribute__((ext_vector_type(2)));
typedef float v8f __attribute__((ext_vector_type(8)));

#define NUM_SRC0 600000
#define NUM_DST0 90000
#define NUM_DST1 8192
#define E0 2250000
#define E1 204800
#define IN_F 100
#define HID 256
#define NCLS 47

// Workspace layout in floats
#define OFF_MSG0 0ull                          // 90000*100  = 9,000,000
#define OFF_DEG0 9000000ull                    // 90000
#define OFF_MSG1 (OFF_DEG0 + 90000ull)         // 8192*256   = 2,097,152
#define OFF_DEG1 (OFF_MSG1 + 2097152ull)       // 8192
#define OFF_H    (OFF_DEG1 + 8192ull)          // 90000*256  = 23,040,000
#define N_ZERO   OFF_H                         // zero everything before h

// ---------------------------------------------------------------- zero
__global__ void zero_kernel(float* __restrict__ p, unsigned long long n) {
    unsigned long long i = (unsigned long long)blockIdx.x * blockDim.x + threadIdx.x;
    unsigned long long stride = (unsigned long long)gridDim.x * blockDim.x;
    for (; i < n; i += stride) p[i] = 0.0f;
}

// ---------------------------------------------------- layer-0 scatter
// one wave32 per edge; lanes stripe the 100 features
__global__ __launch_bounds__(256) void scatter0(const float* __restrict__ x,
                                                const int* __restrict__ src,
                                                const int* __restrict__ dst,
                                                float* __restrict__ msg,
                                                float* __restrict__ deg) {
    int wave = blockIdx.x * 8 + (threadIdx.x >> 5);
    int lane = threadIdx.x & 31;
    int s = src[wave];
    int d = dst[wave];
    const float* xr = x + (unsigned long long)s * IN_F;
    float* mr = msg + (unsigned long long)d * IN_F;
#pragma unroll
    for (int j = 0; j < 3; ++j) {
        int f = lane + j * 32;
        atomicAdd(&mr[f], xr[f]);
    }
    if (lane < 4) {
        int f = lane + 96;
        atomicAdd(&mr[f], xr[f]);
    }
    if (lane == 0) atomicAdd(&deg[d], 1.0f);
}

// ---------------------------------------------------- layer-1 scatter
__global__ __launch_bounds__(256) void scatter1(const float* __restrict__ h,
                                                const int* __restrict__ src,
                                                const int* __restrict__ dst,
                                                float* __restrict__ msg,
                                                float* __restrict__ deg) {
    int wave = blockIdx.x * 8 + (threadIdx.x >> 5);
    int lane = threadIdx.x & 31;
    int s = src[wave];
    int d = dst[wave];
    const float* hr = h + (unsigned long long)s * HID;
    float* mr = msg + (unsigned long long)d * HID;
#pragma unroll
    for (int j = 0; j < 8; ++j) {
        int f = lane + j * 32;
        atomicAdd(&mr[f], hr[f]);
    }
    if (lane == 0) atomicAdd(&deg[d], 1.0f);
}

// ------------------------------------------------------- layer-0 GEMM
// out(90000x256) = relu( x[:90000] @ Ws0 + (msg0/deg) @ Wn0 + b0 )
// one wave per 16x16 tile; V_WMMA_F32_16X16X4_F32, K chunks of 4
__global__ __launch_bounds__(128) void gemm0(const float* __restrict__ x,
                                             const float* __restrict__ Wself,
                                             const float* __restrict__ Wneigh,
                                             const float* __restrict__ bias,
                                             const float* __restrict__ msg,
                                             const float* __restrict__ deg,
                                             float* __restrict__ h) {
    const int N_TILES = HID / 16;  // 16
    int wave = blockIdx.x * 4 + (threadIdx.x >> 5);
    int lane = threadIdx.x & 31;
    int mt = wave / N_TILES;
    int nt = wave % N_TILES;
    int half = lane >> 4;      // 0 or 1
    int l15  = lane & 15;
    int mrow = mt * 16 + l15;  // A-fragment row for this lane
    int ncol = nt * 16 + l15;  // B-fragment col for this lane
    int kh   = half * 2;       // K offset inside 4-chunk: {0,1} or {2,3}

    float invdeg = 1.0f / fmaxf(deg[mrow], 1.0f);
    const float* arow = x   + (unsigned long long)mrow * IN_F;
    const float* nrow = msg + (unsigned long long)mrow * IN_F;

    v8f c = {};
    // self half: A = x rows, B = W_self0 (100x256 row-major)
    for (int k4 = 0; k4 < IN_F; k4 += 4) {
        v2f a, b;
        a.x = arow[k4 + kh];
        a.y = arow[k4 + kh + 1];
        b.x = Wself[(k4 + kh)     * HID + ncol];
        b.y = Wself[(k4 + kh + 1) * HID + ncol];
        c = __builtin_amdgcn_wmma_f32_16x16x4_f32(false, a, false, b,
                                                  (short)0, c, false, false);
    }
    // neighbor half: A = msg/deg rows, B = W_neigh0
    for (int k4 = 0; k4 < IN_F; k4 += 4) {
        v2f a, b;
        a.x = nrow[k4 + kh]     * invdeg;
        a.y = nrow[k4 + kh + 1] * invdeg;
        b.x = Wneigh[(k4 + kh)     * HID + ncol];
        b.y = Wneigh[(k4 + kh + 1) * HID + ncol];
        c = __builtin_amdgcn_wmma_f32_16x16x4_f32(false, a, false, b,
                                                  (short)0, c, false, false);
    }
    float bn = bias[ncol];
#pragma unroll
    for (int r = 0; r < 8; ++r) {
        int m = mt * 16 + r + half * 8;
        float v = c[r] + bn;
        h[(unsigned long long)m * HID + ncol] = fmaxf(v, 0.0f);
    }
}

// ------------------------------------------------------- layer-1 GEMM
// out(8192x47) = h[:8192] @ Ws1 + (msg1/deg) @ Wn1 + b1  (N padded to 48)
__global__ __launch_bounds__(128) void gemm1(const float* __restrict__ h,
                                             const float* __restrict__ Wself,
                                             const float* __restrict__ Wneigh,
                                             const float* __restrict__ bias,
                                             const float* __restrict__ msg,
                                             const float* __restrict__ deg,
                                             float* __restrict__ out) {
    const int N_TILES = 3;  // 48 columns padded
    int wave = blockIdx.x * 4 + (threadIdx.x >> 5);
    int lane = threadIdx.x & 31;
    int mt = wave / N_TILES;
    int nt = wave % N_TILES;
    int half = lane >> 4;
    int l15  = lane & 15;
    int mrow = mt * 16 + l15;
    int ncol = nt * 16 + l15;
    int nvalid = (ncol < NCLS) ? 1 : 0;
    int nc = nvalid ? ncol : 0;        // clamped index -> unconditional loads
    float nmask = nvalid ? 1.0f : 0.0f;
    int kh = half * 2;

    float invdeg = 1.0f / fmaxf(deg[mrow], 1.0f);
    const float* arow = h   + (unsigned long long)mrow * HID;
    const float* nrow = msg + (unsigned long long)mrow * HID;

    v8f c = {};
    for (int k4 = 0; k4 < HID; k4 += 4) {
        v2f a, b;
        a.x = arow[k4 + kh];
        a.y = arow[k4 + kh + 1];
        b.x = Wself[(k4 + kh)     * NCLS + nc] * nmask;
        b.y = Wself[(k4 + kh + 1) * NCLS + nc] * nmask;
        c = __builtin_amdgcn_wmma_f32_16x16x4_f32(false, a, false, b,
                                                  (short)0, c, false, false);
    }
    for (int k4 = 0; k4 < HID; k4 += 4) {
        v2f a, b;
        a.x = nrow[k4 + kh]     * invdeg;
        a.y = nrow[k4 + kh + 1] * invdeg;
        b.x = Wneigh[(k4 + kh)     * NCLS + nc] * nmask;
        b.y = Wneigh[(k4 + kh + 1) * NCLS + nc] * nmask;
        c = __builtin_amdgcn_wmma_f32_16x16x4_f32(false, a, false, b,
                                                  (short)0, c, false, false);
    }
    if (nvalid) {
        float bn = bias[ncol];
#pragma unroll
        for (int r = 0; r < 8; ++r) {
            int m = mt * 16 + r + half * 8;
            out[(unsigned long long)m * NCLS + ncol] = c[r] + bn;
        }
    }
}

// -------------------------------------------------------------- launch
extern "C" void kernel_launch(void* const* d_in, const int* in_sizes, int n_in,
                              void* d_out, int out_size, void* d_ws, size_t ws_size,
                              hipStream_t stream) {
    const float* x    = (const float*)d_in[0];
    const float* Ws0  = (const float*)d_in[1];
    const float* Wn0  = (const float*)d_in[2];
    const float* b0   = (const float*)d_in[3];
    const float* Ws1  = (const float*)d_in[4];
    const float* Wn1  = (const float*)d_in[5];
    const float* b1   = (const float*)d_in[6];
    const int*   src0 = (const int*)d_in[7];
    const int*   dst0 = (const int*)d_in[8];
    const int*   src1 = (const int*)d_in[9];
    const int*   dst1 = (const int*)d_in[10];

    float* ws   = (float*)d_ws;
    float* msg0 = ws + OFF_MSG0;
    float* deg0 = ws + OFF_DEG0;
    float* msg1 = ws + OFF_MSG1;
    float* deg1 = ws + OFF_DEG1;
    float* h    = ws + OFF_H;
    float* out  = (float*)d_out;

    zero_kernel<<<2048, 256, 0, stream>>>(ws, (unsigned long long)N_ZERO);
    scatter0<<<E0 / 8, 256, 0, stream>>>(x, src0, dst0, msg0, deg0);
    gemm0<<<(NUM_DST0 / 16) * (HID / 16) / 4, 128, 0, stream>>>(x, Ws0, Wn0, b0, msg0, deg0, h);
    scatter1<<<E1 / 8, 256, 0, stream>>>(h, src1, dst1, msg1, deg1);
    gemm1<<<(NUM_DST1 / 16) * 3 / 4, 128, 0, stream>>>(h, Ws1, Wn1, b1, msg1, deg1, out);
}